// SingleCellFeaturePredictedGeneExpressionPrior_90623809945964
// MI455X (gfx1250) — compile-verified
//
#include <hip/hip_runtime.h>

// CDNA5 (gfx1250) wave32 kernel for the single-cell gene-expression prior.
//
// Roofline reasoning (MI455X): 0.5 GFLOP total, 256 MB feature stream (HBM,
// touch-once, NT hint), 37.5 MB gathered weight table (kept L2-resident via
// default RT policy; L2 = 192 MB), 12 MB NT output. Memory-bound; VALU kept
// lean with hardware transcendentals (v_exp_f32 / v_log_f32, branch-free).
//
// - Main gathered 64x4 matvec: per-lane VALU FMAs (weights differ per sample).
// - Shared fl0 layer (4->8): V_WMMA_F32_16X16X4_F32 with A = fl0_w padded to
//   16x4, B = y^T (4x16 samples); D[hid, sample] leaves all 8 hidden units of
//   sample j in lane j. Two WMMAs per wave cover its 32 samples.
// - Cross-lane staging via ds_bpermute; no LDS allocation, EXEC stays all-ones
//   through the WMMA region (only cndmask selects before it).

typedef __attribute__((ext_vector_type(2))) float v2f;
typedef __attribute__((ext_vector_type(4))) float v4f;
typedef __attribute__((ext_vector_type(8))) float v8f;

#define GENES 36601
#define NFEAT 64

// Branch-free SELU using the hardware exp (v_exp_f32 via __expf).
__device__ __forceinline__ float seluf(float x) {
  const float scale = 1.0507009873554805f;
  const float alpha = 1.6732632423543772f;
  float e = __expf(fminf(x, 0.f));          // clamp avoids overflow on pos side
  return x > 0.f ? scale * x : scale * alpha * (e - 1.f);
}

__device__ __forceinline__ float log1pfastf(float x) {
  return __logf(1.f + x);                   // v_log_f32, branch-free
}

__device__ __forceinline__ float log_sigmoidf(float x) {
  // -softplus(-x), stable form
  return fminf(x, 0.f) - log1pfastf(__expf(-fabsf(x)));
}

__device__ __forceinline__ float logaddexpf_(float a, float b) {
  float mx = fmaxf(a, b);
  float mn = fminf(a, b);
  return mx + log1pfastf(__expf(mn - mx));
}

__device__ __forceinline__ float bpermf(int byte_idx, float v) {
  return __int_as_float(__builtin_amdgcn_ds_bpermute(byte_idx, __float_as_int(v)));
}

__global__ __launch_bounds__(256) void sc_prior_kernel(
    const int* __restrict__ gidx,            // (N,)
    const float* __restrict__ x,             // (N, 64)
    const float* __restrict__ weight,        // (64, GENES, 4)
    const float* __restrict__ bias,          // (GENES, 4)
    const float* __restrict__ fl0_w,         // (8, 4)
    const float* __restrict__ fl0_b,         // (8,)
    const float* __restrict__ fl1_w,         // (1, 8)
    const float* __restrict__ fl1_b,         // (1,)
    const float* __restrict__ logit_p_conf_g,// (GENES,)
    const float* __restrict__ log_pred_phi_g,// (GENES,)
    const float* __restrict__ prior,         // (GENES, 3)
    float* __restrict__ out,                 // (N, 3)
    int n_total)
{
  const int n    = blockIdx.x * blockDim.x + threadIdx.x;
  const int lane = threadIdx.x & 31;
  const int m    = lane & 15;   // row of A / col of B
  const int hi   = lane >> 4;   // K-pair selector: 0 -> {K0,K1}, 1 -> {K2,K3}

  // Clamp index for the tail block so all loads are safe and EXEC stays full
  // through the WMMA region; only the final store is masked.
  const int ns = (n < n_total) ? n : (n_total - 1);
  const int g  = gidx[ns];

  // ---- Gathered matvec: acc[h] = bias[g,h] + sum_f x[n,f] * W[f,g,h] ----
  // Weights: default (RT) policy -> 37.5 MB table stays L2-resident.
  // Features: nontemporal -> 256 MB stream does not rinse the weight lines.
  const v4f* __restrict__ wq = (const v4f*)weight + g;      // element f at wq[f*GENES]
  const v4f* __restrict__ xq = (const v4f*)(x + (size_t)ns * NFEAT);
  v4f acc = *((const v4f*)bias + g);
#pragma unroll
  for (int k = 0; k < 16; ++k) {
    v4f xv = __builtin_nontemporal_load(xq + k);
    v4f w0 = wq[(size_t)(4 * k + 0) * GENES];
    v4f w1 = wq[(size_t)(4 * k + 1) * GENES];
    v4f w2 = wq[(size_t)(4 * k + 2) * GENES];
    v4f w3 = wq[(size_t)(4 * k + 3) * GENES];
    acc += xv.x * w0 + xv.y * w1 + xv.z * w2 + xv.w * w3;
  }

  float y0 = seluf(acc.x);
  float y1 = seluf(acc.y);
  float y2 = seluf(acc.z);
  float y3 = seluf(acc.w);

  // ---- A matrix: fl0_w zero-padded to 16x4.  A layout (32-bit 16x4):
  //      lane m holds row M=m; v0 = K(hi?2:0), v1 = K(hi?3:1). Rows 8..15 zero. ----
  const int   mi   = (m < 8) ? m : 0;
  const float mask = (m < 8) ? 1.f : 0.f;
  v2f a;
  a.x = fl0_w[mi * 4 + 2 * hi + 0] * mask;
  a.y = fl0_w[mi * 4 + 2 * hi + 1] * mask;

  // ---- B matrices: y^T (4x16).  Lane holds col N=m; v0 = K(hi?2:0), v1 = K(hi?3:1).
  //      Tile0 cols = samples of lanes 0..15, tile1 cols = samples of lanes 16..31. ----
  const int i0 = m * 4;
  const int i1 = (m + 16) * 4;
  v2f b0, b1;
  {
    float s0 = bpermf(i0, y0), s1 = bpermf(i0, y1), s2 = bpermf(i0, y2), s3 = bpermf(i0, y3);
    b0.x = hi ? s2 : s0;
    b0.y = hi ? s3 : s1;
  }
  {
    float s0 = bpermf(i1, y0), s1 = bpermf(i1, y1), s2 = bpermf(i1, y2), s3 = bpermf(i1, y3);
    b1.x = hi ? s2 : s0;
    b1.y = hi ? s3 : s1;
  }

  v8f c = {0.f, 0.f, 0.f, 0.f, 0.f, 0.f, 0.f, 0.f};
  // D[hid, sample]: lane j (0..15) gets hid rows 0..7 for its column's sample in d[0..7].
  v8f d0 = __builtin_amdgcn_wmma_f32_16x16x4_f32(false, a, false, b0, (short)0, c, false, false);
  v8f d1 = __builtin_amdgcn_wmma_f32_16x16x4_f32(false, a, false, b1, (short)0, c, false, false);

  // ---- fl0 bias + SELU + fl1 (8->1), per column lane ----
  float p0 = fl1_b[0];
  float p1 = fl1_b[0];
#pragma unroll
  for (int r = 0; r < 8; ++r) {
    float fb = fl0_b[r];
    float fw = fl1_w[r];
    p0 += seluf(d0[r] + fb) * fw;
    p1 += seluf(d1[r] + fb) * fw;
  }

  // Redistribute: tile1 preds live in lanes 0..15 but belong to samples of lanes 16..31.
  float pm   = bpermf(m * 4, p1);
  float pred = (lane < 16) ? p0 : pm;   // pred for THIS lane's own sample

  // ---- Per-sample log-space mixture math (branch-free hw transcendentals) ----
  float lpphi = log_pred_phi_g[g];
  float lpc   = logit_p_conf_g[g];
  const float* pp = prior + (size_t)g * 3;
  float gmu = pp[0], gphi = pp[1], gz = pp[2];

  float log_gz  = log_sigmoidf(gz);
  float log_gnz = __logf(1.f - __expf(log_gz));
  float log_pc  = log_sigmoidf(lpc);
  float log_pnc = __logf(1.f - __expf(log_pc));

  float num = logaddexpf_(log_pnc + log_gnz + gmu, log_pc + pred);
  float den = logaddexpf_(log_gnz, log_pc + log_gz);
  float log_mu_eff   = num - den;
  float log_phi_eff  = logaddexpf_(log_pc + lpphi, log_pnc + gphi);
  float logit_pz_eff = gz - log1pfastf(__expf(lpc) * (1.f + __expf(gz)));

  if (n < n_total) {
    float* o = out + (size_t)n * 3;
    __builtin_nontemporal_store(log_mu_eff,   o + 0);
    __builtin_nontemporal_store(log_phi_eff,  o + 1);
    __builtin_nontemporal_store(logit_pz_eff, o + 2);
  }
}

extern "C" void kernel_launch(void* const* d_in, const int* in_sizes, int n_in,
                              void* d_out, int out_size, void* d_ws, size_t ws_size,
                              hipStream_t stream) {
  const int*   gidx   = (const int*)d_in[0];
  // d_in[1] = cell_index_tensor_n (unused by reference output)
  const float* x      = (const float*)d_in[2];
  const float* weight = (const float*)d_in[3];
  const float* bias   = (const float*)d_in[4];
  const float* fl0_w  = (const float*)d_in[5];
  const float* fl0_b  = (const float*)d_in[6];
  const float* fl1_w  = (const float*)d_in[7];
  const float* fl1_b  = (const float*)d_in[8];
  const float* lpcg   = (const float*)d_in[9];
  const float* lphig  = (const float*)d_in[10];
  const float* prior  = (const float*)d_in[11];
  float*       out    = (float*)d_out;

  const int n = in_sizes[0];
  const int block = 256;
  const int grid = (n + block - 1) / block;
  sc_prior_kernel<<<grid, block, 0, stream>>>(gidx, x, weight, bias, fl0_w, fl0_b,
                                              fl1_w, fl1_b, lpcg, lphig, prior, out, n);
}